// FCSNN1_53841710023245
// MI455X (gfx1250) — compile-verified
//
#include <hip/hip_runtime.h>

// Problem constants (from reference): T=64, B=512, IN=784, HID=2048, OUT=10
#define TT   64
#define BB   512
#define INN  784
#define HID  2048
#define OUTN 10

#define DT_TAU_MEM 0.1f
#define DT_TAU_SYN 0.2f
#define V_TH       0.5f
#define NEG_INF   (-3.402823466e38f)

typedef __attribute__((ext_vector_type(2))) float v2f;
typedef __attribute__((ext_vector_type(8))) float v8f;

// ---------------------------------------------------------------------------
// Init: zero LIF/LI state, seed output with -inf (output is the running max).
// ---------------------------------------------------------------------------
__global__ __launch_bounds__(256) void snn_init(float* __restrict__ v1,
                                                float* __restrict__ i1,
                                                float* __restrict__ vo,
                                                float* __restrict__ io,
                                                float* __restrict__ out) {
    long long i = (long long)blockIdx.x * blockDim.x + threadIdx.x;
    if (i < (long long)BB * HID) { v1[i] = 0.f; i1[i] = 0.f; }
    if (i < (long long)BB * OUTN) { vo[i] = 0.f; io[i] = 0.f; out[i] = NEG_INF; }
}

// ---------------------------------------------------------------------------
// fc1 GEMM via native fp32 WMMA (V_WMMA_F32_16X16X4_F32).
//   cur[b][h] = sum_k Xt[b][k] * w1[h][k]
// Block: 128 threads = 4 waves; block tile 64(M) x 64(N); each wave: 16x64.
// Fragment layout per ISA 7.12.2:
//   A 16x4 : lane(half,lid) holds A[lid][2*half + {0,1}]     (v2f)
//   B 4x16 : lane(half,lid) holds B[2*half + {0,1}][lid]     (v2f)
//   D 16x16: vgpr r, lane(half,lid) holds D[r + 8*half][lid] (v8f)
// ---------------------------------------------------------------------------
__global__ __launch_bounds__(128) void snn_gemm(const float* __restrict__ x,
                                                const float* __restrict__ w1,
                                                float* __restrict__ cur,
                                                int tBase, long long curStrideT) {
    const int t = tBase + blockIdx.z;
    const float* __restrict__ Xt = x + (long long)t * (BB * INN);
    float* __restrict__ C = cur + (long long)blockIdx.z * curStrideT;

    const int lane = threadIdx.x & 31;
    const int wave = threadIdx.x >> 5;   // 0..3 (M direction)
    const int half = lane >> 4;          // K sub-group {0,1}
    const int lid  = lane & 15;

    const int m  = blockIdx.y * 64 + wave * 16 + lid;  // A row this lane loads
    const int n0 = blockIdx.x * 64 + lid;              // base B column

    const float* __restrict__ arow = Xt + (long long)m * INN + 2 * half;
    const float* __restrict__ b0 = w1 + (long long)(n0 +  0) * INN + 2 * half;
    const float* __restrict__ b1 = w1 + (long long)(n0 + 16) * INN + 2 * half;
    const float* __restrict__ b2 = w1 + (long long)(n0 + 32) * INN + 2 * half;
    const float* __restrict__ b3 = w1 + (long long)(n0 + 48) * INN + 2 * half;

    v8f acc0 = {}, acc1 = {}, acc2 = {}, acc3 = {};

#pragma unroll 4
    for (int k = 0; k < INN; k += 4) {
        v2f a   = *(const v2f*)(arow + k);
        v2f vb0 = *(const v2f*)(b0 + k);
        v2f vb1 = *(const v2f*)(b1 + k);
        v2f vb2 = *(const v2f*)(b2 + k);
        v2f vb3 = *(const v2f*)(b3 + k);
        acc0 = __builtin_amdgcn_wmma_f32_16x16x4_f32(false, a, false, vb0,
                                                     (short)0, acc0, false, false);
        acc1 = __builtin_amdgcn_wmma_f32_16x16x4_f32(false, a, false, vb1,
                                                     (short)0, acc1, false, false);
        acc2 = __builtin_amdgcn_wmma_f32_16x16x4_f32(false, a, false, vb2,
                                                     (short)0, acc2, false, false);
        acc3 = __builtin_amdgcn_wmma_f32_16x16x4_f32(false, a, false, vb3,
                                                     (short)0, acc3, false, false);
    }

    const int rowBase = blockIdx.y * 64 + wave * 16 + half * 8;
    const int colBase = blockIdx.x * 64 + lid;
#pragma unroll
    for (int r = 0; r < 8; ++r) {
        float* crow = C + (long long)(rowBase + r) * HID + colBase;
        crow[0]  = acc0[r];
        crow[16] = acc1[r];
        crow[32] = acc2[r];
        crow[48] = acc3[r];
    }
}

// ---------------------------------------------------------------------------
// One recurrent timestep, fused:
//   LIF update on [B, HID], spike z, readout GEMV z@W_out^T (OUT=10),
//   LI update on [B, OUT], running max into out.
// Block: 256 threads = 4 batch rows x 64 threads; each thread strides 32
// hidden units (coalesced: h = c + 64*j).
// ---------------------------------------------------------------------------
#define ROWS 4
__global__ __launch_bounds__(256) void snn_step(const float* __restrict__ cur,
                                                const float* __restrict__ w_out,
                                                float* __restrict__ v1,
                                                float* __restrict__ i1,
                                                float* __restrict__ vo,
                                                float* __restrict__ io,
                                                float* __restrict__ out) {
    __shared__ float red[ROWS][64][OUTN];

    const int tid = threadIdx.x;
    const int r = tid >> 6;           // 0..3 row in block
    const int c = tid & 63;           // 0..63 column group
    const int b = blockIdx.x * ROWS + r;

    float part[OUTN];
#pragma unroll
    for (int o = 0; o < OUTN; ++o) part[o] = 0.f;

    const long long base = (long long)b * HID;
#pragma unroll 4
    for (int j = 0; j < HID / 64; ++j) {
        const int h = c + 64 * j;
        const long long idx = base + h;
        float v  = v1[idx];
        float i  = i1[idx];
        float cu = cur[idx];
        float vdec = v + DT_TAU_MEM * (-v + i);
        float idec = i * (1.f - DT_TAU_SYN);
        float z = (vdec - V_TH) > 0.f ? 1.f : 0.f;
        v1[idx] = (1.f - z) * vdec;
        i1[idx] = idec + cu;
        if (z != 0.f) {
#pragma unroll
            for (int o = 0; o < OUTN; ++o) part[o] += w_out[o * HID + h];
        }
    }

#pragma unroll
    for (int o = 0; o < OUTN; ++o) red[r][c][o] = part[o];
    __syncthreads();

    if (c < OUTN) {
        float out_cur = 0.f;
        for (int k = 0; k < 64; ++k) out_cur += red[r][k][c];
        const long long bo = (long long)b * OUTN + c;
        float voo = vo[bo];
        float ioo = io[bo];
        float vnew = voo + DT_TAU_MEM * (-voo + ioo);   // uses OLD io
        float inew = ioo * (1.f - DT_TAU_SYN) + out_cur;
        vo[bo] = vnew;
        io[bo] = inew;
        out[bo] = fmaxf(out[bo], vnew);
    }
}

// ---------------------------------------------------------------------------
extern "C" void kernel_launch(void* const* d_in, const int* in_sizes, int n_in,
                              void* d_out, int out_size, void* d_ws, size_t ws_size,
                              hipStream_t stream) {
    (void)in_sizes; (void)n_in; (void)out_size;

    const float* x     = (const float*)d_in[0];  // [T, B, 1, 28, 28] -> [T, B, 784]
    const float* w1    = (const float*)d_in[1];  // [HID, IN]
    const float* w_out = (const float*)d_in[2];  // [OUT, HID]
    float* out = (float*)d_out;                  // [B, OUT] running max

    const size_t smallCur = (size_t)BB * HID;                 // one timestep
    const size_t bigCur   = (size_t)TT * BB * HID;            // all timesteps
    const size_t stateF   = 2 * (size_t)BB * HID + 2 * (size_t)BB * OUTN;

    const bool big = ws_size >= (bigCur + stateF) * sizeof(float);
    const size_t curF = big ? bigCur : smallCur;

    float* cur = (float*)d_ws;
    float* v1  = cur + curF;
    float* i1  = v1 + (size_t)BB * HID;
    float* vo  = i1 + (size_t)BB * HID;
    float* io  = vo + (size_t)BB * OUTN;

    // 1) init state + output running max
    snn_init<<<((size_t)BB * HID + 255) / 256, 256, 0, stream>>>(v1, i1, vo, io, out);

    const dim3 gemmBlk(128, 1, 1);
    const dim3 stepGrid(BB / ROWS, 1, 1);

    if (big) {
        // One batched WMMA GEMM over all timesteps, then the sequential scan.
        dim3 gemmGrid(HID / 64, BB / 64, TT);
        snn_gemm<<<gemmGrid, gemmBlk, 0, stream>>>(x, w1, cur, 0, (long long)BB * HID);
        for (int t = 0; t < TT; ++t) {
            snn_step<<<stepGrid, 256, 0, stream>>>(cur + (size_t)t * BB * HID,
                                                   w_out, v1, i1, vo, io, out);
        }
    } else {
        // Interleave GEMM(t) and step(t) through a single [B,HID] buffer.
        dim3 gemmGrid(HID / 64, BB / 64, 1);
        for (int t = 0; t < TT; ++t) {
            snn_gemm<<<gemmGrid, gemmBlk, 0, stream>>>(x, w1, cur, t, 0);
            snn_step<<<stepGrid, 256, 0, stream>>>(cur, w_out, v1, i1, vo, io, out);
        }
    }
}